// PointNetSetAbstraction_46712064311939
// MI455X (gfx1250) — compile-verified
//
#include <hip/hip_runtime.h>

// ---------------- problem constants ----------------
#define BATCH    16
#define NPTS     4096
#define NPOINT   512
#define NSAMPLE  32
#define R2       0.16f            // 0.4^2
#define C1       64
#define C2       64
#define C3       128
#define CNT_F    262144.0f        // B*NPOINT*NSAMPLE
#define BN_EPS   1e-5f

typedef __attribute__((ext_vector_type(2))) float v2f;
typedef __attribute__((ext_vector_type(8))) float v8f;

// =====================================================================
// K1: farthest point sampling. One workgroup per batch, dist[] in LDS.
// Serial 512 steps; per-step argmax via wave32 shuffles (2 barriers/iter).
// Emits new_xyz (ws) and d_out part 1 (B,3,NPOINT).
// =====================================================================
#define FPS_T 256
__global__ __launch_bounds__(FPS_T)
void fps_kernel(const float* __restrict__ xyz,
                float* __restrict__ newxyz,      // (B*NPOINT,3)
                float* __restrict__ out)         // d_out, part1 (B,3,NPOINT)
{
    __shared__ float dist[NPTS];
    __shared__ float wv[FPS_T / 32];
    __shared__ int   wi[FPS_T / 32];
    __shared__ int   s_far;

    const int b    = blockIdx.x;
    const int t    = threadIdx.x;
    const int lane = t & 31;
    const int wave = t >> 5;
    const float* xb = xyz + (size_t)b * 3 * NPTS;

    for (int j = t; j < NPTS; j += FPS_T) dist[j] = 1e10f;
    if (t == 0) s_far = 0;
    __syncthreads();

    for (int it = 0; it < NPOINT; ++it) {
        const int far = s_far;
        const float cx = xb[far];
        const float cy = xb[NPTS + far];
        const float cz = xb[2 * NPTS + far];
        if (t == 0) {
            float* nz = newxyz + (size_t)(b * NPOINT + it) * 3;
            nz[0] = cx; nz[1] = cy; nz[2] = cz;
            out[(size_t)b * 3 * NPOINT + 0 * NPOINT + it] = cx;
            out[(size_t)b * 3 * NPOINT + 1 * NPOINT + it] = cy;
            out[(size_t)b * 3 * NPOINT + 2 * NPOINT + it] = cz;
        }
        float bestv = -1.0f; int besti = 0;
        for (int j = t; j < NPTS; j += FPS_T) {
            float dx = xb[j] - cx, dy = xb[NPTS + j] - cy, dz = xb[2 * NPTS + j] - cz;
            float d = dx * dx + dy * dy + dz * dz;
            d = fminf(dist[j], d);
            dist[j] = d;
            if (d > bestv) { bestv = d; besti = j; }   // strided order keeps lowest j on ties
        }
        // wave32 argmax reduction (first-index tie-break)
        for (int off = 16; off > 0; off >>= 1) {
            float v2 = __shfl_down(bestv, off, 32);
            int   i2 = __shfl_down(besti, off, 32);
            if (v2 > bestv || (v2 == bestv && i2 < besti)) { bestv = v2; besti = i2; }
        }
        if (lane == 0) { wv[wave] = bestv; wi[wave] = besti; }
        __syncthreads();
        if (t == 0) {
            float bv = wv[0]; int bi = wi[0];
#pragma unroll
            for (int w = 1; w < FPS_T / 32; ++w) {
                if (wv[w] > bv || (wv[w] == bv && wi[w] < bi)) { bv = wv[w]; bi = wi[w]; }
            }
            s_far = bi;
        }
        __syncthreads();
    }
}

// =====================================================================
// K2: ball query. One wave32 per center; ordered ballot-append of first
// NSAMPLE indices with d^2 <= R2; pad with first hit (or 0).
// =====================================================================
__global__ __launch_bounds__(256)
void ballq_kernel(const float* __restrict__ xyz,
                  const float* __restrict__ newxyz,
                  int* __restrict__ gidx)
{
    const int wid  = (int)((blockIdx.x * blockDim.x + threadIdx.x) >> 5);
    const int lane = threadIdx.x & 31;
    if (wid >= BATCH * NPOINT) return;
    const int b = wid >> 9;            // /NPOINT
    const float* xb = xyz + (size_t)b * 3 * NPTS;
    const float cx = newxyz[wid * 3 + 0];
    const float cy = newxyz[wid * 3 + 1];
    const float cz = newxyz[wid * 3 + 2];
    int* outp = gidx + (size_t)wid * NSAMPLE;

    int count = 0;
    int first = 0;
    for (int j0 = 0; j0 < NPTS; j0 += 32) {
        const int j = j0 + lane;
        float dx = xb[j] - cx, dy = xb[NPTS + j] - cy, dz = xb[2 * NPTS + j] - cz;
        float d = dx * dx + dy * dy + dz * dz;
        const bool in = (d <= R2);
        unsigned long long m = __ballot(in);
        if (m) {
            if (count == 0) first = j0 + (__ffsll(m) - 1);
            int pre = __popcll(m & ((1ull << lane) - 1ull));
            if (in && (count + pre) < NSAMPLE) outp[count + pre] = j;
            count += __popcll(m);
            if (count >= NSAMPLE) { count = NSAMPLE; break; }
        }
    }
    // pad remaining slots with the first in-radius index (0 if none)
    if (lane < NSAMPLE && lane >= count) outp[lane] = first;
}

// =====================================================================
// f32 WMMA 16x16x4 micro-tile: D(16x16) += A(16x4) * B(4x16) over kpad.
// A from LDS (row-major, stride ldx); B pre-padded (LDS or global),
// B[k][n] = Bm[n*ldb + k]. No guards -> branch-free, EXEC uniform.
// =====================================================================
__device__ inline v8f wmma_tile(const float* __restrict__ Xs, int ldx,
                                const float* __restrict__ Bm, int ldb,
                                int kpad, int mtile, int ntile, int lane)
{
    v8f acc = {};
    const int kh = (lane >> 4) * 2;       // lanes 0-15: K=k0,k0+1 ; lanes 16-31: K=k0+2,k0+3
    const float* ap = Xs + (mtile * 16 + (lane & 15)) * ldx + kh;
    const float* bp = Bm + (ntile * 16 + (lane & 15)) * ldb + kh;
    for (int k0 = 0; k0 < kpad; k0 += 4) {
        v2f a, bb;
        a.x  = ap[k0];  a.y  = ap[k0 + 1];
        bb.x = bp[k0];  bb.y = bp[k0 + 1];
        acc = __builtin_amdgcn_wmma_f32_16x16x4_f32(false, a, false, bb,
                                                    (short)0, acc, false, false);
    }
    return acc;
}

// =====================================================================
// K3/K5/K7: fused MLP stages. Workgroup = 128 threads (4 waves), 2 groups
// (M = 64 rows). STAGE 0: layer1 stats. STAGE 1: recompute l1, norm, l2
// stats. STAGE 2: recompute l1,l2, norm, l3 stats + per-group max-pool.
// W1 (zero-padded) and W2 staged in LDS; all strides padded for banks.
// =====================================================================
template<int STAGE>
__global__ __launch_bounds__(128)
void mlp_kernel(const float* __restrict__ xyz,  const float* __restrict__ pts,
                const float* __restrict__ newxyz, const int* __restrict__ gidx,
                const float* __restrict__ W1, const float* __restrict__ b1,
                const float* __restrict__ W2, const float* __restrict__ b2,
                const float* __restrict__ W3, const float* __restrict__ b3,
                const float* __restrict__ ss,       // [3][2][128] scale,shift
                float* __restrict__ stats,          // [3][2][128] sum,sumsq
                float* __restrict__ pooled)         // (B*NPOINT, C3) as f32 bits
{
    __shared__ float X0[64 * 12];
    __shared__ float Xa[64 * 68];
    __shared__ float Xb[64 * 68];
    __shared__ float W1s[64 * 12];      // padded to K=8 (cols 6,7 zero)
    __shared__ float W2s[64 * 68];
    __shared__ float s_sum[128];
    __shared__ float s_sq[128];

    const int g0   = blockIdx.x * 2;
    const int t    = threadIdx.x;
    const int wave = t >> 5;
    const int lane = t & 31;

    if (t < 128) { s_sum[t] = 0.0f; s_sq[t] = 0.0f; }

    // ---- stage weights ----
    if (t < 64) {
        float* r = &W1s[t * 12];
#pragma unroll
        for (int k = 0; k < 6; ++k) r[k] = W1[t * 6 + k];
        r[6] = 0.0f; r[7] = 0.0f;
    }
    if (STAGE >= 1) {
        for (int i = t; i < 64 * 64; i += 128)
            W2s[(i >> 6) * 68 + (i & 63)] = W2[i];
    }

    // ---- build grouped features X0 (64 rows x [dx,dy,dz,p0,p1,p2,0,0]) ----
    if (t < 64) {
        const int g   = g0 + (t >> 5);
        const int smp = t & 31;
        const int b   = g >> 9;
        const int idx = gidx[(size_t)g * NSAMPLE + smp];
        const float* xb = xyz + (size_t)b * 3 * NPTS;
        const float* pb = pts + (size_t)b * 3 * NPTS;
        float* r = &X0[t * 12];
        r[0] = xb[idx] - newxyz[g * 3 + 0];
        r[1] = xb[NPTS + idx] - newxyz[g * 3 + 1];
        r[2] = xb[2 * NPTS + idx] - newxyz[g * 3 + 2];
        r[3] = pb[idx];
        r[4] = pb[NPTS + idx];
        r[5] = pb[2 * NPTS + idx];
        r[6] = 0.0f; r[7] = 0.0f;
    }
    __syncthreads();

    // ---------------- layer 1: (64x8) x (8->64) ----------------
    for (int tt = wave; tt < 16; tt += 4) {
        const int mtile = tt >> 2, ntile = tt & 3;
        v8f acc = wmma_tile(X0, 12, W1s, 12, 8, mtile, ntile, lane);
        const int n = ntile * 16 + (lane & 15);
        const float bias = b1[n];
        if (STAGE == 0) {
            float s = 0.0f, q = 0.0f;
#pragma unroll
            for (int i = 0; i < 8; ++i) {
                float v = fmaxf(acc[i] + bias, 0.0f);
                s += v; q += v * v;
            }
            atomicAdd(&s_sum[n], s); atomicAdd(&s_sq[n], q);
        } else {
            const float sc = ss[0 * 256 + n], sh = ss[0 * 256 + 128 + n];
            const int mbase = mtile * 16 + ((lane >> 4) << 3);
#pragma unroll
            for (int i = 0; i < 8; ++i) {
                float v = fmaxf(acc[i] + bias, 0.0f);
                v = fmaxf(sc * v + sh, 0.0f);
                Xa[(mbase + i) * 68 + n] = v;
            }
        }
    }

    if (STAGE >= 1) {
        __syncthreads();
        // ---------------- layer 2: (64x64) x (64->64) ----------------
        for (int tt = wave; tt < 16; tt += 4) {
            const int mtile = tt >> 2, ntile = tt & 3;
            v8f acc = wmma_tile(Xa, 68, W2s, 68, 64, mtile, ntile, lane);
            const int n = ntile * 16 + (lane & 15);
            const float bias = b2[n];
            if (STAGE == 1) {
                float s = 0.0f, q = 0.0f;
#pragma unroll
                for (int i = 0; i < 8; ++i) {
                    float v = fmaxf(acc[i] + bias, 0.0f);
                    s += v; q += v * v;
                }
                atomicAdd(&s_sum[n], s); atomicAdd(&s_sq[n], q);
            } else {
                const float sc = ss[1 * 256 + n], sh = ss[1 * 256 + 128 + n];
                const int mbase = mtile * 16 + ((lane >> 4) << 3);
#pragma unroll
                for (int i = 0; i < 8; ++i) {
                    float v = fmaxf(acc[i] + bias, 0.0f);
                    v = fmaxf(sc * v + sh, 0.0f);
                    Xb[(mbase + i) * 68 + n] = v;
                }
            }
        }
    }

    if (STAGE == 2) {
        __syncthreads();
        // ------------- layer 3: (64x64) x (64->128) + max-pool -------------
        for (int tt = wave; tt < 32; tt += 4) {
            const int mtile = tt >> 3, ntile = tt & 7;   // 4 mtiles x 8 ntiles
            v8f acc = wmma_tile(Xb, 68, W3, 64, 64, mtile, ntile, lane);
            const int n = ntile * 16 + (lane & 15);
            const float bias = b3[n];
            float s = 0.0f, q = 0.0f, vmax = 0.0f;
#pragma unroll
            for (int i = 0; i < 8; ++i) {
                float v = fmaxf(acc[i] + bias, 0.0f);
                s += v; q += v * v;
                vmax = fmaxf(vmax, v);
            }
            atomicAdd(&s_sum[n], s); atomicAdd(&s_sq[n], q);
            const int g = g0 + (mtile >> 1);             // rows 0-31 -> g0, 32-63 -> g0+1
            // post-relu values are >= 0, so int-bit compare == float compare
            atomicMax((int*)&pooled[(size_t)g * C3 + n], __float_as_int(vmax));
        }
    }

    // ---- flush this block's stats for the stage's target layer ----
    __syncthreads();
    const int L = STAGE;
    const int C = (STAGE == 2) ? C3 : C1;
    if (t < C) {
        atomicAdd(&stats[L * 256 + t],       s_sum[t]);
        atomicAdd(&stats[L * 256 + 128 + t], s_sq[t]);
    }
}

// =====================================================================
// K4/K6/K8: fold stats into scale/shift for layer l.
// =====================================================================
__global__ void finalize_kernel(const float* __restrict__ stats,
                                const float* __restrict__ gamma,
                                const float* __restrict__ beta,
                                float* __restrict__ ss, int l, int C)
{
    const int c = threadIdx.x;
    if (c < C) {
        const float mean = stats[l * 256 + c] / CNT_F;
        const float var  = stats[l * 256 + 128 + c] / CNT_F - mean * mean;
        const float sc   = gamma[c] * rsqrtf(var + BN_EPS);
        ss[l * 256 + c]       = sc;
        ss[l * 256 + 128 + c] = beta[c] - mean * sc;
    }
}

// =====================================================================
// K9: apply layer-3 BN+relu to pooled maxima, transpose to (B,C3,NPOINT).
// =====================================================================
__global__ __launch_bounds__(256)
void output_kernel(const float* __restrict__ pooled,
                   const float* __restrict__ ss,
                   float* __restrict__ out)
{
    const int i = blockIdx.x * blockDim.x + threadIdx.x;   // (b*512+s)*128+c
    if (i >= BATCH * NPOINT * C3) return;
    const int c = i & 127;
    const int s = (i >> 7) & (NPOINT - 1);
    const int b = i >> 16;
    float v = pooled[i];
    v = fmaxf(ss[2 * 256 + c] * v + ss[2 * 256 + 128 + c], 0.0f);
    out[(size_t)BATCH * 3 * NPOINT + (size_t)b * C3 * NPOINT + (size_t)c * NPOINT + s] = v;
}

// =====================================================================
// host side
// =====================================================================
extern "C" void kernel_launch(void* const* d_in, const int* in_sizes, int n_in,
                              void* d_out, int out_size, void* d_ws, size_t ws_size,
                              hipStream_t stream)
{
    const float* xyz = (const float*)d_in[0];
    const float* pts = (const float*)d_in[1];
    const float* W1  = (const float*)d_in[2];
    const float* b1  = (const float*)d_in[3];
    const float* g1  = (const float*)d_in[4];
    const float* be1 = (const float*)d_in[5];
    const float* W2  = (const float*)d_in[6];
    const float* b2  = (const float*)d_in[7];
    const float* g2  = (const float*)d_in[8];
    const float* be2 = (const float*)d_in[9];
    const float* W3  = (const float*)d_in[10];
    const float* b3  = (const float*)d_in[11];
    const float* g3  = (const float*)d_in[12];
    const float* be3 = (const float*)d_in[13];

    char* ws = (char*)d_ws;
    size_t off = 0;
    int*   gidx   = (int*)(ws + off);   off += (size_t)BATCH * NPOINT * NSAMPLE * sizeof(int);   // 1 MB
    float* newxyz = (float*)(ws + off); off += (size_t)BATCH * NPOINT * 3 * sizeof(float);       // 96 KB
    float* stats  = (float*)(ws + off); off += 3 * 256 * sizeof(float);                          // 3 KB
    float* ss     = (float*)(ws + off); off += 3 * 256 * sizeof(float);                          // 3 KB
    float* pooled = (float*)(ws + off); off += (size_t)BATCH * NPOINT * C3 * sizeof(float);      // 4 MB

    float* out = (float*)d_out;

    // zero accumulators (captured as memset nodes during graph capture)
    hipMemsetAsync(stats,  0, 3 * 256 * sizeof(float), stream);
    hipMemsetAsync(pooled, 0, (size_t)BATCH * NPOINT * C3 * sizeof(float), stream);

    fps_kernel<<<BATCH, FPS_T, 0, stream>>>(xyz, newxyz, out);

    ballq_kernel<<<(BATCH * NPOINT) / 8, 256, 0, stream>>>(xyz, newxyz, gidx);

    const int nblk = (BATCH * NPOINT) / 2;   // 2 groups per workgroup
    mlp_kernel<0><<<nblk, 128, 0, stream>>>(xyz, pts, newxyz, gidx,
                                            W1, b1, W2, b2, W3, b3, ss, stats, pooled);
    finalize_kernel<<<1, 128, 0, stream>>>(stats, g1, be1, ss, 0, C1);

    mlp_kernel<1><<<nblk, 128, 0, stream>>>(xyz, pts, newxyz, gidx,
                                            W1, b1, W2, b2, W3, b3, ss, stats, pooled);
    finalize_kernel<<<1, 128, 0, stream>>>(stats, g2, be2, ss, 1, C2);

    mlp_kernel<2><<<nblk, 128, 0, stream>>>(xyz, pts, newxyz, gidx,
                                            W1, b1, W2, b2, W3, b3, ss, stats, pooled);
    finalize_kernel<<<1, 128, 0, stream>>>(stats, g3, be3, ss, 2, C3);

    output_kernel<<<(BATCH * NPOINT * C3 + 255) / 256, 256, 0, stream>>>(pooled, ss, out);
}